// SimVQ_83743272337532
// MI455X (gfx1250) — compile-verified
//
#include <hip/hip_runtime.h>
#include <stdint.h>

typedef float v2f __attribute__((ext_vector_type(2)));
typedef float v8f __attribute__((ext_vector_type(8)));

#define N_ROWS 16384   // 16 * 32 * 32
#define K_EMB  4096
#define D_DIM  256
#define B_SZ   16
#define HW     1024
#define ZQ_ELEMS 4194304  // 16*256*32*32

// ---------------- init: reset per-launch state in workspace ----------------
__global__ void k_init(unsigned long long* keys, unsigned int* hist, float* sums) {
    int i = blockIdx.x * blockDim.x + threadIdx.x;
    if (i < N_ROWS) keys[i] = ~0ull;
    if (i < K_EMB)  hist[i] = 0u;
    if (i == 0)     sums[0] = 0.0f;
}

// ---------------- z_e (B,D,H,W) -> flat (N=B*HW, D) via LDS tile transpose --
__global__ void k_transpose(const float* __restrict__ ze, float* __restrict__ flat) {
    __shared__ float tile[32][33];
    int tx = threadIdx.x & 31;      // fast dim within tile
    int ty = threadIdx.x >> 5;      // 0..7
    int p0 = blockIdx.x * 32;       // position tile (HW)
    int d0 = blockIdx.y * 32;       // channel tile
    int b  = blockIdx.z;
    const float* src = ze + (size_t)b * D_DIM * HW;
    #pragma unroll
    for (int j = 0; j < 4; ++j) {
        int d = ty + 8 * j;
        tile[d][tx] = src[(size_t)(d0 + d) * HW + (p0 + tx)];  // coalesced along p
    }
    __syncthreads();
    float* dst = flat + ((size_t)b * HW + p0) * D_DIM + d0;
    #pragma unroll
    for (int j = 0; j < 4; ++j) {
        int p = ty + 8 * j;
        dst[(size_t)p * D_DIM + tx] = tile[tx][p];             // coalesced along d
    }
}

// ---------------- per-row squared L2 norms (one wave32 per row) ------------
__global__ void k_rownorm(const float* __restrict__ mat, float* __restrict__ out, int rows) {
    int wave = (blockIdx.x * blockDim.x + threadIdx.x) >> 5;
    int lane = threadIdx.x & 31;
    if (wave >= rows) return;
    const float* r = mat + (size_t)wave * D_DIM;
    float s = 0.0f;
    #pragma unroll
    for (int j = 0; j < D_DIM / 32; ++j) {
        float v = r[lane + 32 * j];
        s = fmaf(v, v, s);
    }
    #pragma unroll
    for (int o = 16; o > 0; o >>= 1) s += __shfl_xor(s, o, 32);
    if (lane == 0) out[wave] = s;
}

// ---------------- codebook projection: cb[k][d] = emb[k]·proj_w[d] + b[d] --
__global__ void k_project(const float* __restrict__ emb, const float* __restrict__ pw,
                          const float* __restrict__ pb, float* __restrict__ cb) {
    int gid = blockIdx.x * blockDim.x + threadIdx.x;   // k*256 + d
    int k = gid >> 8, d = gid & 255;
    const float* er = emb + (size_t)k * D_DIM;
    const float* wr = pw  + (size_t)d * D_DIM;
    float acc = pb[d];
    #pragma unroll 8
    for (int j = 0; j < D_DIM; ++j) acc = fmaf(er[j], wr[j], acc);
    cb[gid] = acc;
}

// ---------------- WMMA distance GEMM + fused argmin ------------------------
// dist[n][k] = ||flat_n||^2 + ||cb_k||^2 - 2 * flat_n · cb_k
// Register-blocked: each wave computes a 32(M) x 32(N) tile with 4 v8f
// accumulators (4 loads : 4 WMMAs per K-step). 4 waves tile 2x2 -> 64x64
// block so each A/B fragment is shared by 2 waves through the WGP cache.
// Argmin fused: both N-halves min-combined lane-wise, then one 16-lane
// shuffle reduction + one GLOBAL_ATOMIC_MIN_U64 per row-slice.
__global__ void __launch_bounds__(128)
k_dist(const float* __restrict__ flat, const float* __restrict__ cb,
       const float* __restrict__ rown, const float* __restrict__ cbn,
       float* __restrict__ dist, unsigned long long* __restrict__ keys) {
    int lane  = threadIdx.x & 31;
    int wave  = threadIdx.x >> 5;                // 0..3
    int waveM = wave >> 1;                       // 0..1
    int waveN = wave & 1;                        // 0..1
    int row0 = blockIdx.y * 64 + waveM * 32;     // M tile base (flat rows)
    int col0 = blockIdx.x * 64 + waveN * 32;     // N tile base (codebook rows)
    int m  = lane & 15;                          // row-in-tile for A, col for B
    int hi = lane >> 4;                          // 0: K={0,1}  1: K={2,3}

    // A fragments: 16x4 fp32; VGPR0/1 hold K={0,1} (lanes 0-15) / K={2,3} (16-31)
    const v2f* A0 = reinterpret_cast<const v2f*>(flat) + (size_t)(row0 + m)      * (D_DIM / 2) + hi;
    const v2f* A1 = reinterpret_cast<const v2f*>(flat) + (size_t)(row0 + 16 + m) * (D_DIM / 2) + hi;
    // B fragments: 4x16 fp32, lane = column N, same K striping
    const v2f* B0 = reinterpret_cast<const v2f*>(cb) + (size_t)(col0 + m)      * (D_DIM / 2) + hi;
    const v2f* B1 = reinterpret_cast<const v2f*>(cb) + (size_t)(col0 + 16 + m) * (D_DIM / 2) + hi;

    v8f c00 = {}, c01 = {}, c10 = {}, c11 = {};
    #pragma unroll 2
    for (int k2 = 0; k2 < D_DIM / 2; k2 += 2) {  // k2 = k0/2, k0 steps of 4
        v2f a0 = A0[k2];
        v2f a1 = A1[k2];
        v2f b0 = B0[k2];
        v2f b1 = B1[k2];
        c00 = __builtin_amdgcn_wmma_f32_16x16x4_f32(false, a0, false, b0, (short)0, c00, false, false);
        c01 = __builtin_amdgcn_wmma_f32_16x16x4_f32(false, a0, false, b1, (short)0, c01, false, false);
        c10 = __builtin_amdgcn_wmma_f32_16x16x4_f32(false, a1, false, b0, (short)0, c10, false, false);
        c11 = __builtin_amdgcn_wmma_f32_16x16x4_f32(false, a1, false, b1, (short)0, c11, false, false);
    }

    int   colL = col0 + m;                       // left N-half column for this lane
    int   colR = col0 + 16 + m;                  // right N-half column
    float cnL = cbn[colL];
    float cnR = cbn[colR];

    #pragma unroll
    for (int i = 0; i < 2; ++i) {                // row group: c0* then c1*
        v8f cl = i ? c10 : c00;
        v8f cr = i ? c11 : c01;
        #pragma unroll
        for (int v = 0; v < 8; ++v) {
            int r = row0 + 16 * i + v + hi * 8;  // C/D layout: VGPR v, half-wave hi
            float rn = rown[r];
            float dL = rn + cnL - 2.0f * cl[v];
            float dR = rn + cnR - 2.0f * cr[v];
            __builtin_nontemporal_store(dL, &dist[(size_t)r * K_EMB + colL]);
            __builtin_nontemporal_store(dR, &dist[(size_t)r * K_EMB + colR]);
            // order-preserving float -> u32, pack with index
            unsigned int fL = __float_as_uint(dL);
            fL = (fL & 0x80000000u) ? ~fL : (fL | 0x80000000u);
            unsigned int fR = __float_as_uint(dR);
            fR = (fR & 0x80000000u) ? ~fR : (fR | 0x80000000u);
            unsigned long long kL = ((unsigned long long)fL << 32) | (unsigned int)colL;
            unsigned long long kR = ((unsigned long long)fR << 32) | (unsigned int)colR;
            unsigned long long key = (kL < kR) ? kL : kR;   // combine halves lane-wise
            #pragma unroll
            for (int o = 1; o < 16; o <<= 1) {
                unsigned long long other = __shfl_xor(key, o, 32);
                key = (other < key) ? other : key;
            }
            if (m == 0) atomicMin(&keys[r], key);           // GLOBAL_ATOMIC_MIN_U64
        }
    }
}

// ---------------- indices out (as float) + histogram -----------------------
__global__ void k_indices(const unsigned long long* __restrict__ keys,
                          unsigned int* __restrict__ hist, float* __restrict__ out_idx) {
    int n = blockIdx.x * blockDim.x + threadIdx.x;
    if (n >= N_ROWS) return;
    unsigned int idx = (unsigned int)(keys[n] & 0xFFFFFFFFull);
    out_idx[n] = (float)idx;
    atomicAdd(&hist[idx], 1u);
}

// ---------------- z_q gather (straight-through == z_q) + loss reduction ----
__global__ void k_output(const float* __restrict__ ze, const float* __restrict__ cb,
                         const unsigned long long* __restrict__ keys,
                         float* __restrict__ zq_out, float* __restrict__ sum_sq) {
    __shared__ float red[256];
    size_t e = (size_t)blockIdx.x * 256 + threadIdx.x;   // index into (B,D,H,W)
    int b = (int)(e >> 18);          // D*HW = 2^18
    int r = (int)(e & 262143);
    int d = r >> 10;                 // HW = 2^10
    int p = r & 1023;
    int n = b * HW + p;
    unsigned int idx = (unsigned int)(keys[n] & 0xFFFFFFFFull);
    float zq = cb[(size_t)idx * D_DIM + d];
    float diff = zq - ze[e];
    zq_out[e] = zq;                  // fully coalesced
    red[threadIdx.x] = diff * diff;
    __syncthreads();
    #pragma unroll
    for (int s = 128; s > 0; s >>= 1) {
        if (threadIdx.x < (unsigned)s) red[threadIdx.x] += red[threadIdx.x + s];
        __syncthreads();
    }
    if (threadIdx.x == 0) atomicAdd(sum_sq, red[0]);
}

// ---------------- loss + perplexity ----------------------------------------
__global__ void k_final(const unsigned int* __restrict__ hist, const float* __restrict__ sum_sq,
                        float* __restrict__ loss_out, float* __restrict__ ppl_out) {
    __shared__ float red[256];
    float s = 0.0f;
    for (int i = threadIdx.x; i < K_EMB; i += 256) {
        float p = (float)hist[i] / (float)N_ROWS;
        s += p * logf(p + 1e-10f);
    }
    red[threadIdx.x] = s;
    __syncthreads();
    #pragma unroll
    for (int t = 128; t > 0; t >>= 1) {
        if (threadIdx.x < (unsigned)t) red[threadIdx.x] += red[threadIdx.x + t];
        __syncthreads();
    }
    if (threadIdx.x == 0) {
        *ppl_out  = expf(-red[0]);
        // commitment_loss == codebook_loss numerically -> loss = 1.25 * mse
        *loss_out = 1.25f * (*sum_sq) / (float)ZQ_ELEMS;
    }
}

extern "C" void kernel_launch(void* const* d_in, const int* in_sizes, int n_in,
                              void* d_out, int out_size, void* d_ws, size_t ws_size,
                              hipStream_t stream) {
    const float* z_e  = (const float*)d_in[0];   // (16,256,32,32)
    const float* emb  = (const float*)d_in[1];   // (4096,256)
    const float* pw   = (const float*)d_in[2];   // (256,256)
    const float* pb   = (const float*)d_in[3];   // (256,)

    // workspace layout (float units)
    float* ws   = (float*)d_ws;
    float* flat = ws;                                    // 16384*256   = 4194304 f
    float* cb   = ws + 4194304;                          // 4096*256    = 1048576 f
    float* rown = ws + 5242880;                          // 16384 f
    float* cbn  = ws + 5259264;                          // 4096 f
    unsigned long long* keys = (unsigned long long*)(ws + 5263360);  // 16384 u64 (8B aligned)
    unsigned int* hist = (unsigned int*)(ws + 5296128);  // 4096 u32
    float* sums = ws + 5300224;                          // 1 f

    // output layout (flat, return order): z_q_out | loss | perplexity | indices | distances
    float* out_zq   = (float*)d_out;
    float* out_loss = out_zq + ZQ_ELEMS;        // [4194304]
    float* out_ppl  = out_zq + ZQ_ELEMS + 1;    // [4194305]
    float* out_idx  = out_zq + ZQ_ELEMS + 2;    // 16384 entries
    float* out_dist = out_zq + ZQ_ELEMS + 2 + N_ROWS;  // 16384*4096 entries

    k_init<<<64, 256, 0, stream>>>(keys, hist, sums);
    k_transpose<<<dim3(HW / 32, D_DIM / 32, B_SZ), 256, 0, stream>>>(z_e, flat);
    k_rownorm<<<N_ROWS / 8, 256, 0, stream>>>(flat, rown, N_ROWS);
    k_project<<<(K_EMB * D_DIM) / 256, 256, 0, stream>>>(emb, pw, pb, cb);
    k_rownorm<<<K_EMB / 8, 256, 0, stream>>>(cb, cbn, K_EMB);
    k_dist<<<dim3(K_EMB / 64, N_ROWS / 64), 128, 0, stream>>>(flat, cb, rown, cbn, out_dist, keys);
    k_indices<<<N_ROWS / 256, 256, 0, stream>>>(keys, hist, out_idx);
    k_output<<<ZQ_ELEMS / 256, 256, 0, stream>>>(z_e, cb, keys, out_zq, sums);
    k_final<<<1, 256, 0, stream>>>(hist, sums, out_loss, out_ppl);
}